// RNA_RGCN_Advanced_66185446031881
// MI455X (gfx1250) — compile-verified
//
#include <hip/hip_runtime.h>
#include <hip/hip_bf16.h>

#define NN 50000
#define NE 1600000
#define NR 9
#define DIM 128
#define IND 5
#define NG 64

typedef float v2f __attribute__((ext_vector_type(2)));
typedef float v8f __attribute__((ext_vector_type(8)));

// ---------- helpers ----------
__device__ __forceinline__ unsigned fOrd(float f) {
    unsigned u = __float_as_uint(f);
    return (u & 0x80000000u) ? ~u : (u | 0x80000000u);
}
__device__ __forceinline__ float fDec(unsigned u) {
    return (u & 0x80000000u) ? __uint_as_float(u ^ 0x80000000u) : __uint_as_float(~u);
}

// ---------- elementwise ----------
__global__ void k_zero(float* p, int n) {
    int i = blockIdx.x * blockDim.x + threadIdx.x;
    if (i < n) p[i] = 0.f;
}
__global__ void k_relu(float* p, int n) {
    int i = blockIdx.x * blockDim.x + threadIdx.x;
    if (i < n) { float v = p[i]; p[i] = v > 0.f ? v : 0.f; }
}

// ---------- degree count per (dst, relation) ----------
__global__ void k_count(const int* __restrict__ ei, const int* __restrict__ et,
                        float* __restrict__ cnt) {
    int e = blockIdx.x * blockDim.x + threadIdx.x;
    if (e >= NE) return;
    int dst = ei[NE + e];
    int r = et[e];
    unsafeAtomicAdd(&cnt[dst * NR + r], 1.0f);
}

// ---------- layer-1 scatter: sum raw 5-dim features per (dst, rel) ----------
__global__ void k_scatter1(const int* __restrict__ ei, const int* __restrict__ et,
                           const float* __restrict__ x, float* __restrict__ sums1) {
    int e = blockIdx.x * blockDim.x + threadIdx.x;
    if (e >= NE) return;
    int src = ei[e], dst = ei[NE + e], r = et[e];
    const float* xs = &x[src * IND];
    float* s = &sums1[(dst * NR + r) * IND];
#pragma unroll
    for (int d = 0; d < IND; ++d) unsafeAtomicAdd(&s[d], xs[d]);
}

// ---------- layer-1 transform (K=5, scalar VALU is fine) + ReLU ----------
__global__ void k_layer1(const float* __restrict__ x, const float* __restrict__ sums1,
                         const float* __restrict__ cnt, const float* __restrict__ W1,
                         const float* __restrict__ root1, const float* __restrict__ b1,
                         float* __restrict__ h1) {
    int i = blockIdx.x;          // node
    int t = threadIdx.x;         // output dim (128)
    float acc = b1[t];
#pragma unroll
    for (int d = 0; d < IND; ++d) acc += x[i * IND + d] * root1[d * DIM + t];
    for (int r = 0; r < NR; ++r) {
        float c = cnt[i * NR + r]; c = c < 1.f ? 1.f : c;
        float inv = 1.f / c;
#pragma unroll
        for (int d = 0; d < IND; ++d)
            acc += sums1[(i * NR + r) * IND + d] * inv * W1[(r * IND + d) * DIM + t];
    }
    h1[i * DIM + t] = acc > 0.f ? acc : 0.f;
}

// ---------- layer-2 scatter for one relation: agg[dst] += h1[src]/cnt ----------
__global__ void k_scatter2(const int* __restrict__ ei, const int* __restrict__ et,
                           const float* __restrict__ cnt, const float* __restrict__ h1,
                           float* __restrict__ agg, int rel) {
    int gid = blockIdx.x * blockDim.x + threadIdx.x;
    int e = gid >> 5, lane = gid & 31;
    if (e >= NE) return;
    if (et[e] != rel) return;
    int src = ei[e], dst = ei[NE + e];
    float c = cnt[dst * NR + rel]; c = c < 1.f ? 1.f : c;
    float inv = 1.f / c;
    const float4 v = *(const float4*)&h1[src * DIM + lane * 4];
    float* o = &agg[dst * DIM + lane * 4];
    unsafeAtomicAdd(&o[0], v.x * inv);
    unsafeAtomicAdd(&o[1], v.y * inv);
    unsafeAtomicAdd(&o[2], v.z * inv);
    unsafeAtomicAdd(&o[3], v.w * inv);
}

// ---------- fp32 WMMA GEMM: C[M,128] (+)= A[M,128] @ W[128,128] ----------
// grid.x = M/16 ; 256 threads = 8 waves, wave w owns N-tile [16w, 16w+16).
// accumulate=0 : C initialized from bias[n] ; accumulate=1 : C += (reads C).
__global__ void __launch_bounds__(256)
k_gemm_wmma(const float* __restrict__ A, const float* __restrict__ W,
            const float* __restrict__ bias, float* __restrict__ C, int accumulate) {
    __shared__ float As[16 * DIM];           // 8 KB A tile
    const int tid = threadIdx.x;
    const int m0 = blockIdx.x * 16;
#pragma unroll
    for (int i = 0; i < 8; ++i)              // coalesced stage of 16x128 tile
        As[tid + 256 * i] = A[m0 * DIM + tid + 256 * i];
    __syncthreads();

    const int lane = tid & 31;
    const int hi = lane >> 4;                // 0: K pair {0,1}/M rows 0-7 ; 1: {2,3}/rows 8-15
    const int l15 = lane & 15;
    const int n0 = (tid >> 5) * 16;
    const int n = n0 + l15;

    v8f c;
    if (accumulate) {
#pragma unroll
        for (int j = 0; j < 8; ++j) c[j] = C[(m0 + j + 8 * hi) * DIM + n];
    } else {
        float bv = bias[n];
#pragma unroll
        for (int j = 0; j < 8; ++j) c[j] = bv;
    }

    for (int k0 = 0; k0 < DIM; k0 += 4) {
        v2f a, b;
        // A 16x4 layout: lanes0-15 hold K={k0,k0+1}, lanes16-31 K={k0+2,k0+3}, M=l15
        a.x = As[l15 * DIM + k0 + 2 * hi];
        a.y = As[l15 * DIM + k0 + 2 * hi + 1];
        // B 4x16 layout: VGPR0 rows {k0, k0+2}, VGPR1 rows {k0+1, k0+3}, N=l15
        b.x = W[(k0 + 2 * hi) * DIM + n];
        b.y = W[(k0 + 2 * hi + 1) * DIM + n];
        c = __builtin_amdgcn_wmma_f32_16x16x4_f32(false, a, false, b, (short)0, c,
                                                  false, false);
    }
#pragma unroll
    for (int j = 0; j < 8; ++j) C[(m0 + j + 8 * hi) * DIM + n] = c[j];
}

// ---------- attention pooling ----------
__global__ void k_initpool(unsigned* gmax, float* gden) {
    int i = threadIdx.x;
    if (i < NG) { gmax[i] = 0x007FFFFFu /* fOrd(-inf) */; gden[i] = 0.f; }
}
__global__ void k_gate(const float* __restrict__ h2, const float* __restrict__ gw,
                       const float* __restrict__ gb, const int* __restrict__ batch,
                       float* __restrict__ gate, unsigned* __restrict__ gmax) {
    int gid = blockIdx.x * blockDim.x + threadIdx.x;
    int i = gid >> 5, lane = gid & 31;
    if (i >= NN) return;
    float s = 0.f;
#pragma unroll
    for (int t = 0; t < 4; ++t) {
        int d = lane + t * 32;
        s += h2[i * DIM + d] * gw[d];
    }
#pragma unroll
    for (int off = 16; off; off >>= 1) s += __shfl_xor(s, off, 32);
    if (lane == 0) {
        float g = s + gb[0];
        gate[i] = g;
        atomicMax(&gmax[batch[i]], fOrd(g));
    }
}
__global__ void k_expsum(const int* __restrict__ batch, const unsigned* __restrict__ gmax,
                         float* __restrict__ gate, float* __restrict__ gden) {
    int i = blockIdx.x * blockDim.x + threadIdx.x;
    if (i >= NN) return;
    int b = batch[i];
    float e = expf(gate[i] - fDec(gmax[b]));
    gate[i] = e;
    unsafeAtomicAdd(&gden[b], e);
}
__global__ void k_pool(const int* __restrict__ batch, const float* __restrict__ gate,
                       const float* __restrict__ gden, const float* __restrict__ h2,
                       float* __restrict__ pooled) {
    int i = blockIdx.x, d = threadIdx.x;
    int b = batch[i];
    float w = gate[i] / gden[b];
    unsafeAtomicAdd(&pooled[b * DIM + d], w * h2[i * DIM + d]);
}

// ---------- launch ----------
extern "C" void kernel_launch(void* const* d_in, const int* in_sizes, int n_in,
                              void* d_out, int out_size, void* d_ws, size_t ws_size,
                              hipStream_t stream) {
    const float* x      = (const float*)d_in[0];
    const int*   ei     = (const int*)d_in[1];   // [2, NE]
    const int*   et     = (const int*)d_in[2];
    const int*   batch  = (const int*)d_in[3];
    const float* W1     = (const float*)d_in[4]; // [9,5,128]
    const float* root1  = (const float*)d_in[5];
    const float* b1     = (const float*)d_in[6];
    const float* W2     = (const float*)d_in[7]; // [9,128,128]
    const float* root2  = (const float*)d_in[8];
    const float* b2     = (const float*)d_in[9];
    const float* gate_w = (const float*)d_in[10];
    const float* gate_b = (const float*)d_in[11];
    const float* fc_w   = (const float*)d_in[12];
    const float* fc_b   = (const float*)d_in[13];
    float* out = (float*)d_out;

    float* ws     = (float*)d_ws;
    float* cnt    = ws;                          // NN*NR
    float* sums1  = cnt + (size_t)NN * NR;       // NN*NR*IND
    float* h1     = sums1 + (size_t)NN * NR * IND;
    float* agg    = h1 + (size_t)NN * DIM;
    float* h2     = agg + (size_t)NN * DIM;
    float* gate   = h2 + (size_t)NN * DIM;
    unsigned* gmax = (unsigned*)(gate + NN);
    float* gden   = (float*)(gmax + NG);
    float* pooled = gden + NG;

    const int B = 256;
    // degree counts + layer-1 aggregation of raw features
    k_zero<<<(NN * NR + B - 1) / B, B, 0, stream>>>(cnt, NN * NR);
    k_zero<<<(NN * NR * IND + B - 1) / B, B, 0, stream>>>(sums1, NN * NR * IND);
    k_count<<<(NE + B - 1) / B, B, 0, stream>>>(ei, et, cnt);
    k_scatter1<<<(NE + B - 1) / B, B, 0, stream>>>(ei, et, x, sums1);
    k_layer1<<<NN, DIM, 0, stream>>>(x, sums1, cnt, W1, root1, b1, h1);

    // layer 2: h2 = h1@root2 + b2, then += mean_r @ W2[r] per relation
    k_gemm_wmma<<<NN / 16, 256, 0, stream>>>(h1, root2, b2, h2, 0);
    for (int r = 0; r < NR; ++r) {
        k_zero<<<(NN * DIM + B - 1) / B, B, 0, stream>>>(agg, NN * DIM);
        k_scatter2<<<((size_t)NE * 32 + B - 1) / B, B, 0, stream>>>(ei, et, cnt, h1, agg, r);
        k_gemm_wmma<<<NN / 16, 256, 0, stream>>>(agg, W2 + (size_t)r * DIM * DIM, nullptr, h2, 1);
    }
    k_relu<<<(NN * DIM + B - 1) / B, B, 0, stream>>>(h2, NN * DIM);

    // attention pooling (segment softmax) + final FC
    k_initpool<<<1, 64, 0, stream>>>(gmax, gden);
    k_zero<<<(NG * DIM + B - 1) / B, B, 0, stream>>>(pooled, NG * DIM);
    k_gate<<<((size_t)NN * 32 + B - 1) / B, B, 0, stream>>>(h2, gate_w, gate_b, batch, gate, gmax);
    k_expsum<<<(NN + B - 1) / B, B, 0, stream>>>(batch, gmax, gate, gden);
    k_pool<<<NN, DIM, 0, stream>>>(batch, gate, gden, h2, pooled);
    k_gemm_wmma<<<NG / 16, 256, 0, stream>>>(pooled, fc_w, fc_b, out, 0);
}